// CNNFusing_81999515615517
// MI455X (gfx1250) — compile-verified
//
#include <hip/hip_runtime.h>
#include <hip/hip_bf16.h>
#include <math.h>

// ---------------------------------------------------------------------------
// MI455X (gfx1250) implementation.
//  * fp32 WMMA 16x16x4 for all GEMMs (problem is HBM-bound: ~0.7GB moved vs
//    ~117 GFLOPs, so fp32 matrix math is free and bit-exact vs the reference).
//  * Tensor Data Mover (tensor_load_to_lds) double-buffers the weight chunks
//    into LDS, synchronized with s_wait_tensorcnt + workgroup barriers, so
//    the next 32KB weight tile streams in while all 8 waves run WMMA.
// ---------------------------------------------------------------------------

#define HD 256          // hidden size
#define BM 64           // tokens per workgroup in fused kernel
#define KC 32           // K-chunk staged in LDS
#define NCH1 16         // GEMM1 chunks (K = 512)
#define NCH  24         // total chunks  (GEMM1 16 + GEMM2 8)

typedef float v2f __attribute__((ext_vector_type(2)));
typedef float v8f __attribute__((ext_vector_type(8)));
typedef unsigned v4u __attribute__((ext_vector_type(4)));
typedef int v4i __attribute__((ext_vector_type(4)));
typedef int v8i __attribute__((ext_vector_type(8)));

__device__ __forceinline__ v8f wmma4(v2f a, v2f b, v8f c) {
    // D = A(16x4 f32) * B(4x16 f32) + C(16x16 f32)
    return __builtin_amdgcn_wmma_f32_16x16x4_f32(
        /*neg_a=*/false, a, /*neg_b=*/false, b,
        /*c_mod=*/(short)0, c, /*reuse_a=*/false, /*reuse_b=*/false);
}

// TDM: DMA a [KC x 256] f32 tile (row stride 256 elems) from global into LDS.
// D# packing per CDNA5 ISA ch.8: group0 = {count/type/lds_addr/global_addr},
// group1 = {data_size, tensor dims, tile dims, strides}, groups 2/3 unused (2D).
__device__ __forceinline__ void tdm_load_weight_chunk(const float* gsrc, unsigned lds_off) {
    unsigned long long ga = (unsigned long long)(uintptr_t)gsrc;
    v4u g0;
    g0[0] = 1u;                                              // count=1, user mode
    g0[1] = lds_off;                                         // lds_addr (bytes)
    g0[2] = (unsigned)(ga & 0xFFFFFFFFu);                    // global_addr[31:0]
    g0[3] = (unsigned)((ga >> 32) & 0x01FFFFFFu) | (2u << 30); // addr[56:32] | type=2
    v8i g1;
    g1[0] = (int)(2u << 16);          // workgroup_mask=0, data_size=2 (4 bytes)
    g1[1] = (int)(256u << 16);        // tensor_dim0 = 256 (lo16)
    g1[2] = (int)((unsigned)KC << 16);// tensor_dim0 hi=0 | tensor_dim1 = KC
    g1[3] = (int)(256u << 16);        // tensor_dim1 hi=0 | tile_dim0 = 256
    g1[4] = (int)KC;                  // tile_dim1 = KC, tile_dim2 = 0
    g1[5] = (int)256u;                // tensor_dim0_stride = 256 (lo32)
    g1[6] = 0;                        // stride0 hi | stride1 lo
    g1[7] = 0;
    v4i g2 = {0, 0, 0, 0};
    v4i g3 = {0, 0, 0, 0};
#if defined(__clang_major__) && (__clang_major__ >= 23)
    v8i g4 = {0, 0, 0, 0, 0, 0, 0, 0};
    __builtin_amdgcn_tensor_load_to_lds(g0, g1, g2, g3, g4, 0);
#else
    __builtin_amdgcn_tensor_load_to_lds(g0, g1, g2, g3, 0);
#endif
}

// ---------------------------------------------------------------------------
// Kernel 1: exclusive prefix sum of session lengths -> offsets[B+1]
// ---------------------------------------------------------------------------
__global__ __launch_bounds__(256)
void scan_kernel(const int* __restrict__ seq_len, int B, int* __restrict__ offsets) {
    __shared__ int chunk[256];
    const int tid = threadIdx.x;
    const int per = (B + 255) / 256;
    const int base = tid * per;

    int s = 0;
    for (int j = 0; j < per; ++j) {
        int idx = base + j;
        if (idx < B) s += seq_len[idx];
    }
    chunk[tid] = s;
    __syncthreads();
    for (int off = 1; off < 256; off <<= 1) {
        int v = chunk[tid];
        int w = (tid >= off) ? chunk[tid - off] : 0;
        __syncthreads();
        chunk[tid] = v + w;
        __syncthreads();
    }
    int run = (tid == 0) ? 0 : chunk[tid - 1];
    for (int j = 0; j < per; ++j) {
        int idx = base + j;
        if (idx < B) {
            offsets[idx] = run;
            run += seq_len[idx];
        }
    }
    if (tid == 255) offsets[B] = chunk[255];
}

// ---------------------------------------------------------------------------
// Kernel 2: per-session mean of hidden -> means[B][256]
// ---------------------------------------------------------------------------
__global__ __launch_bounds__(256)
void mean_kernel(const float* __restrict__ hidden, const int* __restrict__ offsets,
                 float* __restrict__ means) {
    const int s = blockIdx.x;
    const int ch = threadIdx.x;
    const int lo = offsets[s], hi = offsets[s + 1];
    float acc = 0.f;
    for (int t = lo; t < hi; ++t) acc += hidden[(size_t)t * HD + ch];
    means[(size_t)s * HD + ch] = acc / (float)(hi - lo);
}

// ---------------------------------------------------------------------------
// Kernel 3: M1 = means @ W1 + b1 + b2   (v_mean@W1 is constant per session)
// ---------------------------------------------------------------------------
__global__ __launch_bounds__(256)
void m1_kernel(const float* __restrict__ means, const float* __restrict__ W1,
               const float* __restrict__ b1, const float* __restrict__ b2,
               float* __restrict__ M1) {
    __shared__ float sm[HD];
    const int s = blockIdx.x;
    const int n = threadIdx.x;
    sm[n] = means[(size_t)s * HD + n];
    __syncthreads();
    float acc = b1[n] + b2[n];
    #pragma unroll 8
    for (int k = 0; k < HD; ++k)
        acc = fmaf(sm[k], W1[(size_t)k * HD + n], acc);
    M1[(size_t)s * HD + n] = acc;
}

// ---------------------------------------------------------------------------
// Kernel 4: fused  P = tanh([hidden|pos] @ Wpos + bpos)
//                  G = sigmoid(M1[seg] + P @ W2)
//                  alpha = G @ qw + qb
// 8 waves / block: 4 row-blocks of 16 tokens x 2 column halves of N=256.
// Unified 24-chunk K loop; weight tiles arrive via double-buffered TDM.
// ---------------------------------------------------------------------------
__global__ __launch_bounds__(256)
void fused_alpha_kernel(const float* __restrict__ hidden,
                        const float* __restrict__ pos_table,
                        const float* __restrict__ Wpos,
                        const float* __restrict__ bpos,
                        const float* __restrict__ W2,
                        const float* __restrict__ qw,
                        const float* __restrict__ qb,
                        const float* __restrict__ M1,
                        const int*   __restrict__ rev,
                        const int*   __restrict__ offsets,
                        float* __restrict__ alphaOut,
                        int T, int B) {
    __shared__ float sW[2][KC][HD];     // double-buffered weight chunk (TDM dest)
    __shared__ float sA[BM][KC + 2];    // A chunk (padded: dodge bank conflicts)
    __shared__ float sP[BM][HD + 4];    // P matrix (transpose staging for GEMM2)
    __shared__ float sRed[BM][2];       // alpha partials per column half
    __shared__ int   sSeg[BM];          // session id per local token

    const int tid  = threadIdx.x;
    const int lane = tid & 31;
    const int wid  = tid >> 5;
    const int rowblk  = wid & 3;        // which 16-token block
    const int colhalf = wid >> 2;       // which 128-col half
    const int l16  = lane & 15;
    const int hi16 = lane >> 4;         // ISA: lanes 16-31 hold K+2 / M+8
    const bool leader = (wid == 0);     // TDM issuing wave

    const int blkbase = blockIdx.x * BM;

    // session id per token via binary search on offsets
    if (tid < BM) {
        int t = blkbase + tid;
        int lo = 0, hiB = B;
        while (hiB - lo > 1) {
            int mid = (lo + hiB) >> 1;
            if (offsets[mid] <= t) lo = mid; else hiB = mid;
        }
        sSeg[tid] = lo;
    }

    // accumulators start as bias (C layout: one column per lane)
    v8f acc[8];
    #pragma unroll
    for (int nt = 0; nt < 8; ++nt) {
        float bb = bpos[colhalf * 128 + nt * 16 + l16];
        acc[nt] = (v8f){bb, bb, bb, bb, bb, bb, bb, bb};
    }

    const int arow  = tid >> 2;          // 0..63 : token row staged by this thread
    const int acol0 = (tid & 3) * 8;     // 8 consecutive K per thread
    const int atok  = blkbase + arow;
    const bool avalid = atok < T;
    const int apos  = avalid ? rev[atok] : 0;

    // prime the TDM pipeline with chunk 0
    if (leader)
        tdm_load_weight_chunk(Wpos, (unsigned)(uintptr_t)&sW[0][0][0]);

    for (int c = 0; c < NCH; ++c) {
        // issue chunk c+1 into the buffer freed by chunk c-1's compute
        if (leader && (c + 1) < NCH) {
            const int n1 = c + 1;
            const float* gsrc = (n1 < NCH1) ? (Wpos + (size_t)n1 * KC * HD)
                                            : (W2 + (size_t)(n1 - NCH1) * KC * HD);
            tdm_load_weight_chunk(gsrc, (unsigned)(uintptr_t)&sW[n1 & 1][0][0]);
        }

        if (c < NCH1) {
            // stage A chunk: [hidden | pos_emb] columns kc..kc+KC
            const int kc = c * KC;
            #pragma unroll
            for (int j = 0; j < 8; ++j) {
                int k = kc + acol0 + j;
                float v = 0.f;
                if (avalid)
                    v = (k < HD) ? hidden[(size_t)atok * HD + k]
                                 : pos_table[(size_t)apos * HD + (k - HD)];
                sA[arow][acol0 + j] = v;
            }
        }

        if (c == NCH1) {
            // GEMM1 done: tanh, write P (C layout -> row-major transpose),
            // then re-init accumulators from the per-session M1 table.
            #pragma unroll
            for (int nt = 0; nt < 8; ++nt) {
                const int n = colhalf * 128 + nt * 16 + l16;
                #pragma unroll
                for (int i = 0; i < 8; ++i)
                    sP[rowblk * 16 + i + hi16 * 8][n] = tanhf(acc[nt][i]);
            }
            #pragma unroll
            for (int nt = 0; nt < 8; ++nt) {
                const int n = colhalf * 128 + nt * 16 + l16;
                #pragma unroll
                for (int i = 0; i < 8; ++i)
                    acc[nt][i] = M1[(size_t)sSeg[rowblk * 16 + i + hi16 * 8] * HD + n];
            }
        }

        // chunk c is the oldest outstanding TDM op; c+1 (if any) may still fly
        if (leader) {
            if (c + 1 < NCH) __builtin_amdgcn_s_wait_tensorcnt(1);
            else             __builtin_amdgcn_s_wait_tensorcnt(0);
        }
        __syncthreads();   // publish sW chunk c (+ sA / sP) to all waves

        const int bs = c & 1;
        if (c < NCH1) {
            #pragma unroll
            for (int k = 0; k < KC; k += 4) {
                v2f a = *(const v2f*)&sA[rowblk * 16 + l16][k + 2 * hi16];
                const int kb0 = k + 2 * hi16;
                #pragma unroll
                for (int nt = 0; nt < 8; ++nt) {
                    const int n = colhalf * 128 + nt * 16 + l16;
                    v2f b = { sW[bs][kb0][n], sW[bs][kb0 + 1][n] };
                    acc[nt] = wmma4(a, b, acc[nt]);
                }
            }
        } else {
            const int kcoff = (c - NCH1) * KC;
            #pragma unroll
            for (int k = 0; k < KC; k += 4) {
                v2f a = *(const v2f*)&sP[rowblk * 16 + l16][kcoff + k + 2 * hi16];
                const int kb0 = k + 2 * hi16;
                #pragma unroll
                for (int nt = 0; nt < 8; ++nt) {
                    const int n = colhalf * 128 + nt * 16 + l16;
                    v2f b = { sW[bs][kb0][n], sW[bs][kb0 + 1][n] };
                    acc[nt] = wmma4(a, b, acc[nt]);
                }
            }
        }
        __syncthreads();   // compute done; buffer bs reusable next iteration
    }

    // ---------------- alpha = sigmoid(G) @ qw + qb -------------------------
    float rowsum[8];
    #pragma unroll
    for (int i = 0; i < 8; ++i) rowsum[i] = 0.f;
    #pragma unroll
    for (int nt = 0; nt < 8; ++nt) {
        const int n = colhalf * 128 + nt * 16 + l16;
        const float q = qw[n];
        #pragma unroll
        for (int i = 0; i < 8; ++i) {
            float g = 1.0f / (1.0f + __expf(-acc[nt][i]));
            rowsum[i] = fmaf(g, q, rowsum[i]);
        }
    }
    #pragma unroll
    for (int off = 1; off < 16; off <<= 1) {
        #pragma unroll
        for (int i = 0; i < 8; ++i)
            rowsum[i] += __shfl_xor(rowsum[i], off, 16);
    }
    if (l16 == 0) {
        #pragma unroll
        for (int i = 0; i < 8; ++i)
            sRed[rowblk * 16 + i + hi16 * 8][colhalf] = rowsum[i];
    }
    __syncthreads();
    if (tid < BM) {
        int t = blkbase + tid;
        if (t < T) alphaOut[t] = sRed[tid][0] + sRed[tid][1] + qb[0];
    }
}

// ---------------------------------------------------------------------------
// Kernel 5: h_s[s] = sum_t alpha[t] * hidden[t]   (per session, coalesced)
// ---------------------------------------------------------------------------
__global__ __launch_bounds__(256)
void out_kernel(const float* __restrict__ hidden, const float* __restrict__ alpha,
                const int* __restrict__ offsets, float* __restrict__ out) {
    const int s = blockIdx.x;
    const int ch = threadIdx.x;
    const int lo = offsets[s], hi = offsets[s + 1];
    float acc = 0.f;
    for (int t = lo; t < hi; ++t)
        acc = fmaf(alpha[t], hidden[(size_t)t * HD + ch], acc);
    out[(size_t)s * HD + ch] = acc;
}

// ---------------------------------------------------------------------------
extern "C" void kernel_launch(void* const* d_in, const int* in_sizes, int n_in,
                              void* d_out, int out_size, void* d_ws, size_t ws_size,
                              hipStream_t stream) {
    const float* hidden    = (const float*)d_in[0];
    const float* pos_table = (const float*)d_in[1];
    const float* W_pos     = (const float*)d_in[2];
    const float* b_pos     = (const float*)d_in[3];
    const float* W1        = (const float*)d_in[4];
    const float* b1        = (const float*)d_in[5];
    const float* W2        = (const float*)d_in[6];
    const float* b2        = (const float*)d_in[7];
    const float* qw        = (const float*)d_in[8];
    const float* qb        = (const float*)d_in[9];
    const int*   seq_len   = (const int*)d_in[10];
    const int*   rev       = (const int*)d_in[11];

    const int B = in_sizes[10];
    const int T = in_sizes[11];

    char* w = (char*)d_ws;
    auto align_up = [](size_t v) { return (v + 255) & ~(size_t)255; };
    int*   offsets = (int*)w;              w += align_up((size_t)(B + 1) * sizeof(int));
    float* means   = (float*)w;            w += align_up((size_t)B * HD * sizeof(float));
    float* M1      = (float*)w;            w += align_up((size_t)B * HD * sizeof(float));
    float* alpha   = (float*)w;            w += align_up((size_t)T * sizeof(float));

    scan_kernel<<<1, 256, 0, stream>>>(seq_len, B, offsets);
    mean_kernel<<<B, 256, 0, stream>>>(hidden, offsets, means);
    m1_kernel<<<B, 256, 0, stream>>>(means, W1, b1, b2, M1);
    const int nblk = (T + BM - 1) / BM;
    fused_alpha_kernel<<<nblk, 256, 0, stream>>>(hidden, pos_table, W_pos, b_pos, W2,
                                                 qw, qb, M1, rev, offsets, alpha, T, B);
    out_kernel<<<B, 256, 0, stream>>>(hidden, alpha, offsets, (float*)d_out);
}